// SAGE_23295902614320
// MI455X (gfx1250) — compile-verified
//
#include <hip/hip_runtime.h>
#include <math.h>

// ---------------- problem constants (match reference) ----------------
#define N_NODES 50000
#define N_EDGES 800000
#define D_RAW   7
#define DIM     256
#define K_CODES 1024

typedef __attribute__((ext_vector_type(2))) float v2f;
typedef __attribute__((ext_vector_type(8))) float v8f;

// ---------------- output layout (floats, concatenated in return order) ----
#define OUT_H    0
#define OUT_Q    (N_NODES * DIM)                       // 12,800,000
#define OUT_IND  (2 * N_NODES * DIM)                   // 25,600,000
#define OUT_LOSS (2 * N_NODES * DIM + N_NODES)         // 25,650,000
#define OUT_DIST (2 * N_NODES * DIM + N_NODES + 1)     // 25,650,001

// ---------------- workspace layout (floats) ----------------
#define WS_DEGO 0                        // 50000   -> becomes inv_sqrt_out
#define WS_DEGI (WS_DEGO + N_NODES)      // 50000   -> becomes inv_sqrt_in
#define WS_AGG  (WS_DEGI + N_NODES)      // 12.8M
#define WS_H0   (WS_AGG + N_NODES * DIM) // 12.8M
#define WS_X2   (WS_H0 + N_NODES * DIM)  // 50000
#define WS_C2   (WS_X2 + N_NODES)        // 1024
#define WS_ACC  (WS_C2 + K_CODES)        // 1

// =====================================================================
// 0. zero-fill
// =====================================================================
__global__ void k_zero(float* __restrict__ p, int n) {
    int i = blockIdx.x * blockDim.x + threadIdx.x;
    int stride = gridDim.x * blockDim.x;
    for (; i < n; i += stride) p[i] = 0.0f;
}

// =====================================================================
// 1. h0 = feats @ W2 + b2      [N,7] @ [7,256]
//    one block per node; feats row staged through LDS
// =====================================================================
__global__ void k_linear2(const float* __restrict__ feats,
                          const float* __restrict__ W2,
                          const float* __restrict__ b2,
                          float* __restrict__ h0) {
    __shared__ float f[D_RAW];
    const int n = blockIdx.x;
    const int d = threadIdx.x;
    if (d < D_RAW) f[d] = feats[n * D_RAW + d];
    __syncthreads();
    float acc = b2[d];
#pragma unroll
    for (int k = 0; k < D_RAW; ++k) acc += f[k] * W2[k * DIM + d];
    h0[(size_t)n * DIM + d] = acc;
}

// =====================================================================
// 2. degrees via atomics (float accumulation like segment_sum of ones)
// =====================================================================
__global__ void k_degree(const int* __restrict__ esrc, const int* __restrict__ edst,
                         float* __restrict__ deg_out, float* __restrict__ deg_in) {
    int e = blockIdx.x * blockDim.x + threadIdx.x;
    if (e >= N_EDGES) return;
    atomicAdd(&deg_out[esrc[e]], 1.0f);
    atomicAdd(&deg_in[edst[e]], 1.0f);
}

// deg -> rsqrt(deg + 1)  (self-loop) applied to both arrays (contiguous)
__global__ void k_invsqrt(float* __restrict__ deg, int n) {
    int i = blockIdx.x * blockDim.x + threadIdx.x;
    if (i < n) deg[i] = rsqrtf(deg[i] + 1.0f);
}

// =====================================================================
// 3. edge scatter: agg[dst] += h0[src] * iso[src]
//    4 edges per 256-thread block; 64 lanes x float4 per edge
// =====================================================================
__global__ void k_scatter(const int* __restrict__ esrc, const int* __restrict__ edst,
                          const float* __restrict__ h0, const float* __restrict__ iso,
                          float* __restrict__ agg) {
    const int e = blockIdx.x * 4 + (threadIdx.x >> 6);
    const int q = (threadIdx.x & 63) << 2;     // feature offset, float4 granularity
    __builtin_prefetch(&esrc[e + 256], 0, 0);  // global_prefetch_b8 flavor
    const int src = esrc[e];
    const int dst = edst[e];
    const float s = iso[src];
    const float4 hv = *reinterpret_cast<const float4*>(h0 + (size_t)src * DIM + q);
    float* a = agg + (size_t)dst * DIM + q;
    atomicAdd(a + 0, hv.x * s);
    atomicAdd(a + 1, hv.y * s);
    atomicAdd(a + 2, hv.z * s);
    atomicAdd(a + 3, hv.w * s);
}

// =====================================================================
// 4. agg = (agg + h0 * iso) * isi     (self-loop + in-normalization)
// =====================================================================
__global__ void k_norm(float* __restrict__ agg, const float* __restrict__ h0,
                       const float* __restrict__ iso, const float* __restrict__ isi) {
    const int n = blockIdx.x;
    const int d = threadIdx.x;
    const size_t idx = (size_t)n * DIM + d;
    agg[idx] = (agg[idx] + h0[idx] * iso[n]) * isi[n];
}

// =====================================================================
// 5. H = relu(A @ Wg + bg)   via V_WMMA_F32_16X16X4_F32
//    block = 256 thr (8 waves); wave w owns 16x16 tile at (m0, n0+16w)
//    A fragment (16x4 f32): lanes 0-15 -> K=k,k+1 ; lanes 16-31 -> K=k+2,k+3
//    B fragment (4x16 f32): same K split, N = lane%16
//    C/D: VGPR v -> rows m0+v (lanes 0-15) / m0+v+8 (lanes 16-31), N = lane%16
// =====================================================================
__global__ void k_gemm_h(const float* __restrict__ A,   // [N_NODES, DIM] normalized
                         const float* __restrict__ Wg,  // [DIM, DIM] row-major
                         const float* __restrict__ bg,  // [DIM]
                         float* __restrict__ H) {       // [N_NODES, DIM]
    const int lane = threadIdx.x & 31;
    const int wave = threadIdx.x >> 5;
    const int half = lane >> 4;
    const int l16  = lane & 15;
    const int m0 = blockIdx.y * 16;
    const int n0 = blockIdx.x * 128 + wave * 16;

    v8f c = {};
    const float* arow = A + (size_t)(m0 + l16) * DIM;
    const int col = n0 + l16;
    for (int k = 0; k < DIM; k += 4) {
        const int kk = k + half * 2;
        v2f a = *reinterpret_cast<const v2f*>(arow + kk);        // A[m][kk..kk+1]
        v2f b;
        b.x = Wg[(size_t)kk * DIM + col];                        // B[kk][n]
        b.y = Wg[(size_t)(kk + 1) * DIM + col];                  // B[kk+1][n]
        c = __builtin_amdgcn_wmma_f32_16x16x4_f32(false, a, false, b,
                                                  (short)0, c, false, false);
    }
    const float bias = bg[col];
#pragma unroll
    for (int v = 0; v < 8; ++v) {
        const int row = m0 + v + half * 8;
        const float val = c[v] + bias;
        H[(size_t)row * DIM + col] = val > 0.0f ? val : 0.0f;
    }
}

// =====================================================================
// 6. row sum-of-squares (for x2 and c2): one block per row, LDS tree
// =====================================================================
__global__ void k_rowsumsq(const float* __restrict__ X, float* __restrict__ out) {
    __shared__ float s[DIM];
    const int r = blockIdx.x;
    const int t = threadIdx.x;
    const float v = X[(size_t)r * DIM + t];
    s[t] = v * v;
    __syncthreads();
    for (int w = DIM / 2; w > 0; w >>= 1) {
        if (t < w) s[t] += s[t + w];
        __syncthreads();
    }
    if (t == 0) out[r] = s[0];
}

// =====================================================================
// 7. dist = x2 + c2 - 2 * (X @ CB^T)   via V_WMMA_F32_16X16X4_F32
//    B[k][n] = CB[n][k]  -> per-lane contiguous float2 loads from CB rows
// =====================================================================
__global__ void k_gemm_dist(const float* __restrict__ X,    // [N_NODES, DIM]
                            const float* __restrict__ CB,   // [K_CODES, DIM]
                            const float* __restrict__ x2,   // [N_NODES]
                            const float* __restrict__ c2,   // [K_CODES]
                            float* __restrict__ dist) {     // [N_NODES, K_CODES]
    const int lane = threadIdx.x & 31;
    const int wave = threadIdx.x >> 5;
    const int half = lane >> 4;
    const int l16  = lane & 15;
    const int m0 = blockIdx.y * 16;
    const int n0 = blockIdx.x * 128 + wave * 16;

    v8f c = {};
    const float* arow  = X + (size_t)(m0 + l16) * DIM;
    const float* cbrow = CB + (size_t)(n0 + l16) * DIM;
    for (int k = 0; k < DIM; k += 4) {
        const int kk = k + half * 2;
        v2f a = *reinterpret_cast<const v2f*>(arow + kk);
        v2f b = *reinterpret_cast<const v2f*>(cbrow + kk);
        c = __builtin_amdgcn_wmma_f32_16x16x4_f32(false, a, false, b,
                                                  (short)0, c, false, false);
    }
    const int col = n0 + l16;
    const float cc = c2[col];
#pragma unroll
    for (int v = 0; v < 8; ++v) {
        const int row = m0 + v + half * 8;
        dist[(size_t)row * K_CODES + col] = x2[row] + cc - 2.0f * c[v];
    }
}

// =====================================================================
// 8. per-node argmin over dist row, quantize gather, commit-loss partial
// =====================================================================
__global__ void k_argmin_loss(const float* __restrict__ dist,
                              const float* __restrict__ X,
                              const float* __restrict__ CB,
                              float* __restrict__ ind_out,
                              float* __restrict__ quant_out,
                              float* __restrict__ acc) {
    __shared__ float sval[DIM];
    __shared__ int   sidx[DIM];
    __shared__ float ssum[DIM];
    const int n = blockIdx.x;
    const int t = threadIdx.x;

    float best = 3.4e38f;
    int bi = 0;
    const float* drow = dist + (size_t)n * K_CODES;
#pragma unroll
    for (int j = t; j < K_CODES; j += DIM) {
        const float v = drow[j];
        if (v < best) { best = v; bi = j; }
    }
    sval[t] = best; sidx[t] = bi;
    __syncthreads();
    for (int w = DIM / 2; w > 0; w >>= 1) {
        if (t < w) {
            const float ov = sval[t + w]; const int oi = sidx[t + w];
            if (ov < sval[t] || (ov == sval[t] && oi < sidx[t])) {
                sval[t] = ov; sidx[t] = oi;
            }
        }
        __syncthreads();
    }
    const int ind = sidx[0];
    if (t == 0) ind_out[n] = (float)ind;

    const float x = X[(size_t)n * DIM + t];
    const float q = CB[(size_t)ind * DIM + t];
    quant_out[(size_t)n * DIM + t] = q;   // straight-through forward value
    const float dq = q - x;
    ssum[t] = dq * dq;
    __syncthreads();
    for (int w = DIM / 2; w > 0; w >>= 1) {
        if (t < w) ssum[t] += ssum[t + w];
        __syncthreads();
    }
    if (t == 0) atomicAdd(acc, ssum[0]);
}

__global__ void k_loss_final(const float* __restrict__ acc, float* __restrict__ loss) {
    loss[0] = acc[0] * (1.0f / ((float)N_NODES * (float)DIM));
}

// =====================================================================
// launcher
// =====================================================================
extern "C" void kernel_launch(void* const* d_in, const int* in_sizes, int n_in,
                              void* d_out, int out_size, void* d_ws, size_t ws_size,
                              hipStream_t stream) {
    const float* feats = (const float*)d_in[0];
    const float* W2    = (const float*)d_in[1];
    const float* b2    = (const float*)d_in[2];
    const float* Wg    = (const float*)d_in[3];
    const float* bg    = (const float*)d_in[4];
    const float* CB    = (const float*)d_in[5];
    const int*   esrc  = (const int*)d_in[6];
    const int*   edst  = (const int*)d_in[7];

    float* out = (float*)d_out;
    float* ws  = (float*)d_ws;

    float* deg_out = ws + WS_DEGO;   // -> inv_sqrt_out after k_invsqrt
    float* deg_in  = ws + WS_DEGI;   // -> inv_sqrt_in
    float* agg     = ws + WS_AGG;
    float* h0      = ws + WS_H0;
    float* x2      = ws + WS_X2;
    float* c2      = ws + WS_C2;
    float* acc     = ws + WS_ACC;

    float* H    = out + OUT_H;
    float* Q    = out + OUT_Q;
    float* IND  = out + OUT_IND;
    float* LOSS = out + OUT_LOSS;
    float* DIST = out + OUT_DIST;

    // 0. zero: degrees + agg (contiguous WS_DEGO..WS_AGG end) and the loss acc
    k_zero<<<2048, 256, 0, stream>>>(ws, WS_AGG + N_NODES * DIM);
    k_zero<<<1, 32, 0, stream>>>(acc, 1);

    // 1. linear_2
    k_linear2<<<N_NODES, DIM, 0, stream>>>(feats, W2, b2, h0);

    // 2. degrees -> inv sqrt
    k_degree<<<(N_EDGES + 255) / 256, 256, 0, stream>>>(esrc, edst, deg_out, deg_in);
    k_invsqrt<<<(2 * N_NODES + 255) / 256, 256, 0, stream>>>(deg_out, 2 * N_NODES);

    // 3. edge scatter-add
    k_scatter<<<N_EDGES / 4, 256, 0, stream>>>(esrc, edst, h0, deg_out, agg);

    // 4. self-loop + in-normalization
    k_norm<<<N_NODES, DIM, 0, stream>>>(agg, h0, deg_out, deg_in);

    // 5. GraphConv GEMM + bias + relu (WMMA f32)
    k_gemm_h<<<dim3(DIM / 128, N_NODES / 16), 256, 0, stream>>>(agg, Wg, bg, H);

    // 6. squared norms
    k_rowsumsq<<<N_NODES, DIM, 0, stream>>>(H, x2);
    k_rowsumsq<<<K_CODES, DIM, 0, stream>>>(CB, c2);

    // 7. distance GEMM (WMMA f32)
    k_gemm_dist<<<dim3(K_CODES / 128, N_NODES / 16), 256, 0, stream>>>(H, CB, x2, c2, DIST);

    // 8. argmin + quantize + commitment loss
    k_argmin_loss<<<N_NODES, DIM, 0, stream>>>(DIST, H, CB, IND, Q, acc);
    k_loss_final<<<1, 1, 0, stream>>>(acc, LOSS);
}